// GCN_1666447311118
// MI455X (gfx1250) — compile-verified
//
#include <hip/hip_runtime.h>
#include <hip/hip_bf16.h>

typedef __attribute__((ext_vector_type(2))) float v2f;
typedef __attribute__((ext_vector_type(8))) float v8f;

#define HID 128
#define WSTRIDE 132   // padded LDS stride: 132 % 64 == 4 -> conflict-free b64 reads

// ---------------- utility ----------------
__global__ void k_zero_i(int* __restrict__ p, int n) {
  int i = blockIdx.x * blockDim.x + threadIdx.x;
  if (i < n) p[i] = 0;
}
__global__ void k_zero_f(float* __restrict__ p, int n) {
  int i = blockIdx.x * blockDim.x + threadIdx.x;
  if (i < n) p[i] = 0.0f;
}

// ---------------- CSR build: histogram, scan, fill ----------------
__global__ void k_count(const int* __restrict__ dst, int* __restrict__ cnt, int e) {
  int i = blockIdx.x * blockDim.x + threadIdx.x;
  if (i < e) atomicAdd(&cnt[dst[i]], 1);
}

__global__ void k_dinv(const int* __restrict__ cnt, float* __restrict__ dinv, int n) {
  int i = blockIdx.x * blockDim.x + threadIdx.x;
  if (i < n) dinv[i] = rsqrtf((float)cnt[i] + 1.0f);   // +1 self-loop; always > 0
}

// phase 1: per-1024-block inclusive scan; rowstart[i+1] = local inclusive
__global__ void k_scan1(const int* __restrict__ counts, int* __restrict__ rowstart,
                        int* __restrict__ blockSums, int n) {
  __shared__ int sm[1024];
  int t = threadIdx.x;
  int i = blockIdx.x * 1024 + t;
  int v = (i < n) ? counts[i] : 0;
  sm[t] = v;
  __syncthreads();
  for (int off = 1; off < 1024; off <<= 1) {
    int u = (t >= off) ? sm[t - off] : 0;
    __syncthreads();
    sm[t] += u;
    __syncthreads();
  }
  if (i < n) rowstart[i + 1] = sm[t];
  if (t == 1023) blockSums[blockIdx.x] = sm[1023];
}

// phase 2: single-block exclusive scan of block sums (nb <= 1024)
__global__ void k_scan2(const int* __restrict__ blockSums, int* __restrict__ blockOff, int nb) {
  __shared__ int sm[1024];
  int t = threadIdx.x;
  int v = (t < nb) ? blockSums[t] : 0;
  sm[t] = v;
  __syncthreads();
  for (int off = 1; off < 1024; off <<= 1) {
    int u = (t >= off) ? sm[t - off] : 0;
    __syncthreads();
    sm[t] += u;
    __syncthreads();
  }
  if (t < nb) blockOff[t] = sm[t] - v;   // exclusive
}

// phase 3: add block offsets; set rowstart[0]
__global__ void k_scan3(int* __restrict__ rowstart, const int* __restrict__ blockOff, int n) {
  int i = blockIdx.x * blockDim.x + threadIdx.x;
  if (i < n) rowstart[i + 1] += blockOff[i >> 10];
  if (i == 0) rowstart[0] = 0;
}

// fill (src, weight) records grouped by dst
__global__ void k_fill(const int* __restrict__ src, const int* __restrict__ dst,
                       const float* __restrict__ dinv, const int* __restrict__ rowstart,
                       int* __restrict__ cursor, int2* __restrict__ rec, int e) {
  int i = blockIdx.x * blockDim.x + threadIdx.x;
  if (i >= e) return;
  int s = src[i], d = dst[i];
  int p = atomicAdd(&cursor[d], 1);
  int2 r;
  r.x = s;
  r.y = __float_as_int(dinv[s] * dinv[d]);
  rec[rowstart[d] + p] = r;
}

// ---------------- H = act(X + bias) @ W, f32 WMMA ----------------
// Block: 256 threads (8 waves); block computes 128 rows x 128 cols; wave = 16 rows.
// If bias != nullptr, applies (x + bias) then ReLU to the A fragments (fused prev-layer epilogue).
__global__ void k_gemm128(const float* __restrict__ X, const float* __restrict__ W,
                          const float* __restrict__ bias, float* __restrict__ H, int n) {
  extern __shared__ float ws[];               // W^T staged: [128 cols][WSTRIDE]
  const int tid = threadIdx.x;

  for (int i = tid; i < HID * HID; i += 256) {
    int k = i >> 7, c = i & 127;
    ws[c * WSTRIDE + k] = W[i];
  }
  __syncthreads();

  const int wave   = tid >> 5;
  const int lane   = tid & 31;
  const int lane15 = lane & 15;
  const int half   = lane >> 4;
  const int rowBase = blockIdx.x * 128 + wave * 16;
  const int arow    = rowBase + lane15;       // A: lane holds row M
  const int koff    = half * 2;               // half-wave selects K pair

  v2f a[32];
  if (arow < n) {
    const float* xp = X + (size_t)arow * HID + koff;
    if (bias) {
      const float* bp = bias + koff;
#pragma unroll
      for (int kk = 0; kk < 32; ++kk) {
        v2f xv = *(const v2f*)(xp + kk * 4);
        v2f bv = *(const v2f*)(bp + kk * 4);
        v2f t = xv + bv;
        t.x = fmaxf(t.x, 0.0f);
        t.y = fmaxf(t.y, 0.0f);
        a[kk] = t;
      }
    } else {
#pragma unroll
      for (int kk = 0; kk < 32; ++kk)
        a[kk] = *(const v2f*)(xp + kk * 4);
    }
  } else {
    v2f z = {};
#pragma unroll
    for (int kk = 0; kk < 32; ++kk) a[kk] = z;
  }

#pragma unroll
  for (int nt = 0; nt < 8; ++nt) {
    v8f acc = {};
    const float* bp = ws + (nt * 16 + lane15) * WSTRIDE + koff;  // B: lane holds col N
#pragma unroll
    for (int kk = 0; kk < 32; ++kk) {
      v2f b = *(const v2f*)(bp + kk * 4);
      acc = __builtin_amdgcn_wmma_f32_16x16x4_f32(
          false, a[kk], false, b, (short)0, acc, false, false);
    }
    const int colOut = nt * 16 + lane15;
#pragma unroll
    for (int r = 0; r < 8; ++r) {             // D: VGPR r -> M = r + half*8
      int row = rowBase + r + half * 8;
      if (row < n) H[(size_t)row * HID + colOut] = acc[r];
    }
  }
}

// ---------------- AGG[n] = dinv[n]^2*H[n] + sum_in-edges w*H[src]  (no atomics) ----------------
// one wave per node; lane owns 4 features; per-edge record fetched as scalar s_load
__global__ void k_aggregate(const float* __restrict__ H, const int2* __restrict__ rec,
                            const int* __restrict__ rowstart, const float* __restrict__ dinv,
                            float* __restrict__ AGG, int n) {
  long long gid = (long long)blockIdx.x * blockDim.x + threadIdx.x;
  int node = (int)(gid >> 5);
  if (node >= n) return;
  int q = ((int)gid & 31) * 4;

  int beg = __builtin_amdgcn_readfirstlane(rowstart[node]);     // -> SGPR
  int end = __builtin_amdgcn_readfirstlane(rowstart[node + 1]); // -> SGPR

  float wself = dinv[node];
  wself *= wself;
  float4 acc = *(const float4*)(H + (size_t)node * HID + q);
  acc.x *= wself; acc.y *= wself; acc.z *= wself; acc.w *= wself;

  for (int j = beg; j < end; ++j) {           // j uniform -> rec[j] is a scalar load
    int2 r = rec[j];
    int s = r.x;
    float w = __int_as_float(r.y);
    const float4 h = *(const float4*)(H + (size_t)s * HID + q); // coalesced 512B gather
    acc.x += w * h.x; acc.y += w * h.y; acc.z += w * h.z; acc.w += w * h.w;
  }
  *(float4*)(AGG + (size_t)node * HID + q) = acc;
}

// ---------------- pooling ----------------
__global__ void k_cnt(const int* __restrict__ batch, float* __restrict__ cnt, int n) {
  int i = blockIdx.x * blockDim.x + threadIdx.x;
  if (i < n)
    __hip_atomic_fetch_add(&cnt[batch[i]], 1.0f, __ATOMIC_RELAXED, __HIP_MEMORY_SCOPE_AGENT);
}

__global__ void k_pool(const float* __restrict__ H, const int* __restrict__ batch,
                       float* __restrict__ pooled, int n) {
  long long gid = (long long)blockIdx.x * blockDim.x + threadIdx.x;
  int node = (int)(gid >> 5);
  if (node >= n) return;
  int q = ((int)gid & 31) * 4;
  int g = __builtin_amdgcn_readfirstlane(batch[node]);
  const float4 h = *(const float4*)(H + (size_t)node * HID + q);
  float* p = pooled + (size_t)g * HID + q;
  __hip_atomic_fetch_add(p + 0, h.x, __ATOMIC_RELAXED, __HIP_MEMORY_SCOPE_AGENT);
  __hip_atomic_fetch_add(p + 1, h.y, __ATOMIC_RELAXED, __HIP_MEMORY_SCOPE_AGENT);
  __hip_atomic_fetch_add(p + 2, h.z, __ATOMIC_RELAXED, __HIP_MEMORY_SCOPE_AGENT);
  __hip_atomic_fetch_add(p + 3, h.w, __ATOMIC_RELAXED, __HIP_MEMORY_SCOPE_AGENT);
}

// ---------------- out[g][c] = (pooled[g]/max(cnt,1) + b3) . Wl[:,c] + bl[c] ----------------
__global__ void k_final(const float* __restrict__ pooled, const float* __restrict__ cnt,
                        const float* __restrict__ b3, const float* __restrict__ Wl,
                        const float* __restrict__ bl, float* __restrict__ out, int G, int C) {
  int gid = blockIdx.x * blockDim.x + threadIdx.x;
  if (gid >= G * C) return;
  int g = gid / C, c = gid % C;
  float inv = 1.0f / fmaxf(cnt[g], 1.0f);
  float s = 0.0f;
  const float* pr = pooled + (size_t)g * HID;
  for (int k = 0; k < HID; ++k)
    s += (pr[k] * inv + b3[k]) * Wl[k * C + c];   // layer-3 bias folded in
  out[gid] = s + bl[c];
}

// =========================================================================
extern "C" void kernel_launch(void* const* d_in, const int* in_sizes, int n_in,
                              void* d_out, int out_size, void* d_ws, size_t ws_size,
                              hipStream_t stream) {
  const float* x   = (const float*)d_in[0];
  const int*   ei  = (const int*)d_in[1];
  const int*   bat = (const int*)d_in[2];
  const float* W1  = (const float*)d_in[3];
  const float* b1  = (const float*)d_in[4];
  const float* W2  = (const float*)d_in[5];
  const float* b2  = (const float*)d_in[6];
  const float* W3  = (const float*)d_in[7];
  const float* b3  = (const float*)d_in[8];
  const float* Wl  = (const float*)d_in[9];
  const float* bl  = (const float*)d_in[10];

  const int n = in_sizes[2];          // N_NODES
  const int e = in_sizes[1] / 2;      // N_EDGES
  const int C = in_sizes[10];         // N_CLASSES
  const int G = out_size / C;         // N_GRAPHS
  const int* src = ei;
  const int* dst = ei + e;

  // workspace carve (256B-aligned chunks)
  char* base = (char*)d_ws;
  auto alloc = [&](size_t bytes) -> void* {
    void* p = (void*)base;
    base += (bytes + 255) & ~(size_t)255;
    return p;
  };
  int*   counts    = (int*)alloc((size_t)n * 4);
  int*   cursor    = (int*)alloc((size_t)n * 4);
  int*   rowstart  = (int*)alloc(((size_t)n + 1) * 4);
  int*   blockSums = (int*)alloc(1024 * 4);
  int*   blockOff  = (int*)alloc(1024 * 4);
  float* dinv      = (float*)alloc((size_t)n * 4);
  int2*  rec       = (int2*)alloc((size_t)e * 8);
  float* Hb        = (float*)alloc((size_t)n * HID * 4);
  float* Ab        = (float*)alloc((size_t)n * HID * 4);
  float* pooled    = (float*)alloc((size_t)G * HID * 4);
  float* cnt       = (float*)alloc((size_t)G * 4);
  (void)ws_size; (void)n_in;

  const int T = 256;
  auto nb = [](long long w, int t) { return (unsigned)((w + t - 1) / t); };
  const long long nw = (long long)n * 32;               // node-waves (4 feats/lane)
  const unsigned gemmB = (unsigned)((n + 127) / 128);
  const unsigned scanB = (unsigned)((n + 1023) / 1024); // <= 1024 blocks
  const size_t lds = (size_t)HID * WSTRIDE * sizeof(float);  // 67,584 B

  // ---- CSR build + norms (once) ----
  k_zero_i<<<nb(n, T), T, 0, stream>>>(counts, n);
  k_zero_i<<<nb(n, T), T, 0, stream>>>(cursor, n);
  k_count<<<nb(e, T), T, 0, stream>>>(dst, counts, e);
  k_dinv<<<nb(n, T), T, 0, stream>>>(counts, dinv, n);
  k_scan1<<<scanB, 1024, 0, stream>>>(counts, rowstart, blockSums, n);
  k_scan2<<<1, 1024, 0, stream>>>(blockSums, blockOff, (int)scanB);
  k_scan3<<<nb(n, T), T, 0, stream>>>(rowstart, blockOff, n);
  k_fill<<<nb(e, T), T, 0, stream>>>(src, dst, dinv, rowstart, cursor, rec, e);

  // ---- layer 1: x -> Ab ----
  k_gemm128<<<gemmB, T, lds, stream>>>(x, W1, nullptr, Hb, n);
  k_aggregate<<<nb(nw, T), T, 0, stream>>>(Hb, rec, rowstart, dinv, Ab, n);

  // ---- layer 2: relu(Ab+b1) -> Ab (bias+relu fused into GEMM A-load) ----
  k_gemm128<<<gemmB, T, lds, stream>>>(Ab, W2, b1, Hb, n);
  k_aggregate<<<nb(nw, T), T, 0, stream>>>(Hb, rec, rowstart, dinv, Ab, n);

  // ---- layer 3: relu(Ab+b2) -> Ab ----
  k_gemm128<<<gemmB, T, lds, stream>>>(Ab, W3, b2, Hb, n);
  k_aggregate<<<nb(nw, T), T, 0, stream>>>(Hb, rec, rowstart, dinv, Ab, n);

  // ---- mean pool + classifier (b3 folded into k_final) ----
  k_zero_f<<<nb((long long)G * HID, T), T, 0, stream>>>(pooled, G * HID);
  k_zero_f<<<nb(G, T), T, 0, stream>>>(cnt, G);
  k_cnt<<<nb(n, T), T, 0, stream>>>(bat, cnt, n);
  k_pool<<<nb(nw, T), T, 0, stream>>>(Ab, bat, pooled, n);
  k_final<<<nb((long long)G * C, T), T, 0, stream>>>(pooled, cnt, b3, Wl, bl,
                                                     (float*)d_out, G, C);
}